// MultiHeadAttentionLayer_79499844649020
// MI455X (gfx1250) — compile-verified
//
#include <hip/hip_runtime.h>

typedef __attribute__((ext_vector_type(2))) float v2f;
typedef __attribute__((ext_vector_type(8))) float v8f;

#define NNODES 20000
#define NEDGES 200000
#define NTOK   8
#define CIN    64
#define COUT   64          // N_HEADS * HEAD_DIM
#define NHEADS 4

// D = A(16x4) * B(4x16) + C, fp32, one wave
static __device__ __forceinline__ v8f wmma4(v2f a, v2f b, v8f c) {
    return __builtin_amdgcn_wmma_f32_16x16x4_f32(
        /*neg_a=*/false, a, /*neg_b=*/false, b,
        /*c_mod=*/(short)0, c, /*reuse_a=*/false, /*reuse_b=*/false);
}

// Load all 16 A-fragments (one 16x64 fp32 tile, rows row0..row0+15).
// ISA 32-bit A 16x4 layout: lane<16 -> {K=4k, 4k+1}; lane>=16 -> {K=4k+2, 4k+3}
static __device__ __forceinline__ void load_a(const float* __restrict__ base,
                                              int row0, int l16, int half,
                                              v2f a[16]) {
    const float* p = base + (size_t)(row0 + l16) * CIN + 2 * half;
#pragma unroll
    for (int k = 0; k < 16; ++k)
        a[k] = *(const v2f*)(p + 4 * k);
}

// One 16x16 output tile of X(16x64) @ W(64x64) + bias, columns [16t, 16t+16)
// B (4x16) layout mirrors A: v0 = W[4k+2*half][ncol], v1 = W[4k+2*half+1][ncol]
static __device__ __forceinline__ v8f gemm_tile(const v2f a[16],
                                                const float* __restrict__ W,
                                                float bias, int ncol, int half) {
    v8f c;
#pragma unroll
    for (int r = 0; r < 8; ++r) c[r] = bias;
#pragma unroll
    for (int k = 0; k < 16; ++k) {
        const float* wp = W + (size_t)(4 * k + 2 * half) * COUT + ncol;
        v2f b;
        b.x = wp[0];
        b.y = wp[COUT];
        c = wmma4(a[k], b, c);
    }
    return c;
}

// ---------------- node projections: Qh/Kh/Vh = h @ W + b ----------------
// grid = (NNODES*NTOK)/16 blocks of 128 threads; wave w owns N-tile w.
__global__ void node_proj(const float* __restrict__ h,
                          const float* __restrict__ Qw, const float* __restrict__ Qb,
                          const float* __restrict__ Kw, const float* __restrict__ Kb,
                          const float* __restrict__ Vw, const float* __restrict__ Vb,
                          float* __restrict__ Qh, float* __restrict__ Kh,
                          float* __restrict__ Vh) {
    const int lane = threadIdx.x & 31;
    const int wv   = threadIdx.x >> 5;   // n-tile 0..3
    const int half = lane >> 4;
    const int l16  = lane & 15;
    const int tile = blockIdx.x;
    const int row0 = tile * 16;
    const int ncol = wv * 16 + l16;

    v2f a[16];
    load_a(h, row0, l16, half, a);

    // C/D layout: VGPR r -> row (r + 8*half), col = ncol
    {
        v8f c = gemm_tile(a, Qw, Qb[ncol], ncol, half);
        float* o = Qh + (size_t)row0 * COUT;
#pragma unroll
        for (int r = 0; r < 8; ++r) o[(size_t)(r + 8 * half) * COUT + ncol] = c[r];
    }
    {
        v8f c = gemm_tile(a, Kw, Kb[ncol], ncol, half);
        float* o = Kh + (size_t)row0 * COUT;
#pragma unroll
        for (int r = 0; r < 8; ++r) o[(size_t)(r + 8 * half) * COUT + ncol] = c[r];
    }
    {
        v8f c = gemm_tile(a, Vw, Vb[ncol], ncol, half);
        float* o = Vh + (size_t)row0 * COUT;
#pragma unroll
        for (int r = 0; r < 8; ++r) o[(size_t)(r + 8 * half) * COUT + ncol] = c[r];
    }
}

// ---------------- fused edge kernel ----------------
// One wave handles 2 edges (16 rows). Lane half <-> edge, C-VGPR r <-> token,
// N-tile t <-> head (HEAD_DIM == 16).
// grid = NEDGES/16 blocks of 256 threads (8 waves, 16 edges per block).
__global__ void edge_attn(const float* __restrict__ e,
                          const int* __restrict__ src, const int* __restrict__ dst,
                          const float* __restrict__ Ew, const float* __restrict__ Eb,
                          const float* __restrict__ Qh, const float* __restrict__ Kh,
                          const float* __restrict__ Vh,
                          float* __restrict__ e_out,
                          float* __restrict__ wV, float* __restrict__ z) {
    const int lane = threadIdx.x & 31;
    const int wv   = threadIdx.x >> 5;
    const int gw   = blockIdx.x * 8 + wv;     // global wave id, 0..99999
    const int half = lane >> 4;
    const int l16  = lane & 15;
    const int edge = 2 * gw + half;           // this half-wave's edge

    const int s_node = src[edge];
    const int d_node = dst[edge];

    v2f a[16];
    load_a(e, gw * 16, l16, half, a);         // rows = 2 edges x 8 tokens

#pragma unroll
    for (int t = 0; t < NHEADS; ++t) {
        const int ncol = t * 16 + l16;        // col = head*16 + d
        v8f pe = gemm_tile(a, Ew, Eb[ncol], ncol, half);

        const size_t kbase = ((size_t)s_node * NTOK) * COUT + ncol; // Kh/Vh gather
        const size_t qbase = ((size_t)d_node * NTOK) * COUT + ncol; // Qh gather
        const size_t obase = ((size_t)edge * NTOK) * COUT + ncol;   // e_out
        const size_t wbase = ((size_t)d_node * NTOK) * COUT + ncol; // wV scatter

#pragma unroll
        for (int r = 0; r < NTOK; ++r) {
            float kv = Kh[kbase + (size_t)r * COUT];
            float qv = Qh[qbase + (size_t)r * COUT];
            float sc = kv * qv * 0.25f + pe[r];   // 1/sqrt(16) = 0.25
            e_out[obase + (size_t)r * COUT] = sc;

            // head-dim (16-lane) butterfly reduction
            float sum = sc;
#pragma unroll
            for (int m = 1; m < 16; m <<= 1)
                sum += __shfl_xor(sum, m, 32);
            float s = __expf(fminf(fmaxf(sum, -5.0f), 5.0f));

            float vvv = Vh[kbase + (size_t)r * COUT];
            atomicAdd(&wV[wbase + (size_t)r * COUT], vvv * s);
            if (l16 == 0)
                atomicAdd(&z[((size_t)d_node * NTOK + r) * NHEADS + t], s);
        }
    }
}

// ---------------- h_out = wV / (z + 1e-6) ----------------
__global__ void norm_out(const float* __restrict__ wV, const float* __restrict__ z,
                         float* __restrict__ h_out) {
    const size_t total = (size_t)NNODES * NTOK * COUT;
    size_t i = (size_t)blockIdx.x * blockDim.x + threadIdx.x;
    if (i >= total) return;
    int col    = (int)(i & (COUT - 1));
    int head   = col >> 4;
    size_t row = i >> 6;                      // node*NTOK + tok
    float zz = z[row * NHEADS + head];
    h_out[i] = wV[i] / (zz + 1e-6f);
}

extern "C" void kernel_launch(void* const* d_in, const int* in_sizes, int n_in,
                              void* d_out, int out_size, void* d_ws, size_t ws_size,
                              hipStream_t stream) {
    const float* h   = (const float*)d_in[0];
    const float* e   = (const float*)d_in[1];
    const int*   src = (const int*)d_in[2];
    const int*   dst = (const int*)d_in[3];
    const float* Qw  = (const float*)d_in[4];
    const float* Qb  = (const float*)d_in[5];
    const float* Kw  = (const float*)d_in[6];
    const float* Kb  = (const float*)d_in[7];
    const float* Vw  = (const float*)d_in[8];
    const float* Vb  = (const float*)d_in[9];
    const float* Ew  = (const float*)d_in[10];
    const float* Eb  = (const float*)d_in[11];

    const size_t projN = (size_t)NNODES * NTOK * COUT;   // 10,240,000
    float* h_out = (float*)d_out;
    float* e_out = h_out + projN;

    float* Qh = (float*)d_ws;
    float* Kh = Qh + projN;
    float* Vh = Kh + projN;
    float* wV = Vh + projN;
    float* zz = wV + projN;                  // NNODES*NTOK*NHEADS floats

    // zero the accumulators (capture-safe)
    hipMemsetAsync(wV, 0,
                   (projN + (size_t)NNODES * NTOK * NHEADS) * sizeof(float),
                   stream);

    node_proj<<<(NNODES * NTOK) / 16, 128, 0, stream>>>(
        h, Qw, Qb, Kw, Kb, Vw, Vb, Qh, Kh, Vh);

    edge_attn<<<NEDGES / 16, 256, 0, stream>>>(
        e, src, dst, Ew, Eb, Qh, Kh, Vh, e_out, wV, zz);

    const size_t total = projN;
    norm_out<<<(unsigned)((total + 255) / 256), 256, 0, stream>>>(wV, zz, h_out);
}